// myGCN_87308095193621
// MI455X (gfx1250) — compile-verified
//
#include <hip/hip_runtime.h>

typedef float v2f __attribute__((ext_vector_type(2)));
typedef float v8f __attribute__((ext_vector_type(8)));

#define N_NODES   50000
#define N_EDGES   800000
#define HID       256

__device__ __forceinline__ float celu1(float v) {
    return v > 0.0f ? v : expm1f(v);
}

// ---------------------------------------------------------------------------
// C[M x 256] = A[M x 256] @ B[256 x 256]  (+ optional bias + celu)
// One wave computes a 16x64 tile with 4 f32 WMMA accumulators.
// block = 128 threads (4 waves); grid = (ceil(mtiles/4), 4 ntiles)
// ---------------------------------------------------------------------------
template <bool FUSE>
__global__ void gemm256_wmma(const float* __restrict__ A,
                             const float* __restrict__ B,
                             const float* __restrict__ bias,
                             float* __restrict__ C, int mtiles) {
    const int lane  = threadIdx.x & 31;
    const int wave  = threadIdx.x >> 5;
    const int mtile = blockIdx.x * 4 + wave;
    if (mtile >= mtiles) return;                 // whole-wave uniform exit
    const int ntile = blockIdx.y;                // 0..3 -> 64 columns each
    const int half  = lane >> 4;                 // 0: K{k,k+1}, 1: K{k+2,k+3}
    const int l15   = lane & 15;
    const int m     = mtile * 16 + l15;
    const int n0    = ntile * 64;

    v8f acc0 = (v8f)0.0f, acc1 = (v8f)0.0f, acc2 = (v8f)0.0f, acc3 = (v8f)0.0f;

    const float* arow = A + (size_t)m * HID;
#pragma unroll 4
    for (int k0 = 0; k0 < HID; k0 += 4) {
        const int kb = k0 + 2 * half;
        v2f a;
        a.x = arow[kb];
        a.y = arow[kb + 1];
        const float* b0 = B + (size_t)kb * HID;
        const float* b1 = B + (size_t)(kb + 1) * HID;
        {
            v2f b; const int n = n0 + 0 * 16 + l15; b.x = b0[n]; b.y = b1[n];
            acc0 = __builtin_amdgcn_wmma_f32_16x16x4_f32(false, a, false, b,
                                                         (short)0, acc0, false, false);
        }
        {
            v2f b; const int n = n0 + 1 * 16 + l15; b.x = b0[n]; b.y = b1[n];
            acc1 = __builtin_amdgcn_wmma_f32_16x16x4_f32(false, a, false, b,
                                                         (short)0, acc1, false, false);
        }
        {
            v2f b; const int n = n0 + 2 * 16 + l15; b.x = b0[n]; b.y = b1[n];
            acc2 = __builtin_amdgcn_wmma_f32_16x16x4_f32(false, a, false, b,
                                                         (short)0, acc2, false, false);
        }
        {
            v2f b; const int n = n0 + 3 * 16 + l15; b.x = b0[n]; b.y = b1[n];
            acc3 = __builtin_amdgcn_wmma_f32_16x16x4_f32(false, a, false, b,
                                                         (short)0, acc3, false, false);
        }
    }

    v8f accs[4] = {acc0, acc1, acc2, acc3};
#pragma unroll
    for (int t = 0; t < 4; ++t) {
        const int n = n0 + t * 16 + l15;
        const float bv = FUSE ? bias[n] : 0.0f;
#pragma unroll
        for (int i = 0; i < 8; ++i) {
            const int row = mtile * 16 + i + 8 * half;   // C/D layout: VGPR i -> M=i / i+8
            float v = accs[t][i];
            if (FUSE) v = celu1(v + bv);
            C[(size_t)row * HID + n] = v;
        }
    }
}

// ---------------------------------------------------------------------------
// Helpers: zeroing, degree, dinv
// ---------------------------------------------------------------------------
__global__ void zero_f4(float* __restrict__ p, int n4) {
    int i = blockIdx.x * blockDim.x + threadIdx.x;
    if (i < n4) ((float4*)p)[i] = make_float4(0.f, 0.f, 0.f, 0.f);
}

__global__ void deg_count(const long long* __restrict__ dst,
                          float* __restrict__ deg, int E) {
    int e = blockIdx.x * blockDim.x + threadIdx.x;
    if (e < E) atomicAdd(&deg[(int)dst[e]], 1.0f);
}

__global__ void finish_dinv(float* __restrict__ deg, int n) {
    int i = blockIdx.x * blockDim.x + threadIdx.x;
    if (i < n) deg[i] = rsqrtf(deg[i] + 1.0f);
}

// ---------------------------------------------------------------------------
// Edge scatter: agg[dst] += h[src] * dinv[src]*dinv[dst]; one wave per edge
// ---------------------------------------------------------------------------
__global__ void edge_scatter(const float* __restrict__ h,
                             const long long* __restrict__ src,
                             const long long* __restrict__ dst,
                             const float* __restrict__ dinv,
                             float* __restrict__ agg, int E) {
    int e = blockIdx.x * 8 + (threadIdx.x >> 5);
    if (e >= E) return;
    const int lane = threadIdx.x & 31;
    const int s = (int)src[e];
    const int d = (int)dst[e];
    const float nrm = dinv[s] * dinv[d];
    const float4* hs = (const float4*)(h + (size_t)s * HID);
    float* ad = agg + (size_t)d * HID;
#pragma unroll
    for (int j = 0; j < 2; ++j) {
        const int idx = lane + j * 32;
        float4 hv = hs[idx];
        const int base = idx * 4;
        atomicAdd(ad + base + 0, hv.x * nrm);
        atomicAdd(ad + base + 1, hv.y * nrm);
        atomicAdd(ad + base + 2, hv.z * nrm);
        atomicAdd(ad + base + 3, hv.w * nrm);
    }
}

// ---------------------------------------------------------------------------
// Node epilogue (conv1): x = celu(agg + h*dinv^2 + bias); wave per node
// ---------------------------------------------------------------------------
__global__ void node_update(const float* __restrict__ agg,
                            const float* __restrict__ h,
                            const float* __restrict__ dinv,
                            const float* __restrict__ bias,
                            float* __restrict__ xout, int n) {
    int node = blockIdx.x * 8 + (threadIdx.x >> 5);
    if (node >= n) return;
    const int lane = threadIdx.x & 31;
    float d2 = dinv[node]; d2 *= d2;
    const float4* av = (const float4*)(agg + (size_t)node * HID);
    const float4* hv = (const float4*)(h + (size_t)node * HID);
    const float4* bv = (const float4*)bias;
    float4* xo = (float4*)(xout + (size_t)node * HID);
#pragma unroll
    for (int j = 0; j < 2; ++j) {
        const int idx = lane + j * 32;
        float4 a = av[idx], hh = hv[idx], b = bv[idx], r;
        r.x = celu1(a.x + hh.x * d2 + b.x);
        r.y = celu1(a.y + hh.y * d2 + b.y);
        r.z = celu1(a.z + hh.z * d2 + b.z);
        r.w = celu1(a.w + hh.w * d2 + b.w);
        xo[idx] = r;
    }
}

// ---------------------------------------------------------------------------
// Node epilogue (hidden convs): x = LN(celu(agg + h*dinv^2 + bias)) * g + b
// wave per node; wave32 shuffle reductions over the 256 features
// ---------------------------------------------------------------------------
__global__ void node_update_ln(const float* __restrict__ agg,
                               const float* __restrict__ h,
                               const float* __restrict__ dinv,
                               const float* __restrict__ bias,
                               const float* __restrict__ g,
                               const float* __restrict__ bln,
                               float* __restrict__ xout, int n) {
    int node = blockIdx.x * 8 + (threadIdx.x >> 5);
    if (node >= n) return;
    const int lane = threadIdx.x & 31;
    float d2 = dinv[node]; d2 *= d2;
    const float4* av = (const float4*)(agg + (size_t)node * HID);
    const float4* hv = (const float4*)(h + (size_t)node * HID);
    const float4* bv = (const float4*)bias;
    float v[8];
    float s = 0.0f;
#pragma unroll
    for (int j = 0; j < 2; ++j) {
        const int idx = lane + j * 32;
        float4 a = av[idx], hh = hv[idx], b = bv[idx];
        v[j * 4 + 0] = celu1(a.x + hh.x * d2 + b.x);
        v[j * 4 + 1] = celu1(a.y + hh.y * d2 + b.y);
        v[j * 4 + 2] = celu1(a.z + hh.z * d2 + b.z);
        v[j * 4 + 3] = celu1(a.w + hh.w * d2 + b.w);
        s += v[j * 4 + 0] + v[j * 4 + 1] + v[j * 4 + 2] + v[j * 4 + 3];
    }
#pragma unroll
    for (int off = 16; off > 0; off >>= 1) s += __shfl_xor(s, off, 32);
    const float mu = s * (1.0f / 256.0f);
    float ss = 0.0f;
#pragma unroll
    for (int i = 0; i < 8; ++i) { float dd = v[i] - mu; ss += dd * dd; }
#pragma unroll
    for (int off = 16; off > 0; off >>= 1) ss += __shfl_xor(ss, off, 32);
    const float inv = rsqrtf(ss * (1.0f / 256.0f) + 1e-5f);

    const float4* gv = (const float4*)g;
    const float4* lb = (const float4*)bln;
    float4* xo = (float4*)(xout + (size_t)node * HID);
#pragma unroll
    for (int j = 0; j < 2; ++j) {
        const int idx = lane + j * 32;
        float4 gg = gv[idx], bb = lb[idx], r;
        r.x = (v[j * 4 + 0] - mu) * inv * gg.x + bb.x;
        r.y = (v[j * 4 + 1] - mu) * inv * gg.y + bb.y;
        r.z = (v[j * 4 + 2] - mu) * inv * gg.z + bb.z;
        r.w = (v[j * 4 + 3] - mu) * inv * gg.w + bb.w;
        xo[idx] = r;
    }
}

// ---------------------------------------------------------------------------
// Final: out[n][c] = sigmoid(x[n] . W[:,c] + b[c]),  c < 25
// ---------------------------------------------------------------------------
__global__ void final_lin(const float* __restrict__ x,
                          const float* __restrict__ W,
                          const float* __restrict__ b,
                          float* __restrict__ out, int n) {
    int t = blockIdx.x * blockDim.x + threadIdx.x;
    if (t >= n * 25) return;
    const int node = t / 25, c = t % 25;
    const float* xr = x + (size_t)node * HID;
    float acc = b[c];
#pragma unroll 8
    for (int k = 0; k < HID; ++k) acc += xr[k] * W[k * 25 + c];
    out[t] = 1.0f / (1.0f + expf(-acc));
}

// ---------------------------------------------------------------------------
extern "C" void kernel_launch(void* const* d_in, const int* in_sizes, int n_in,
                              void* d_out, int out_size, void* d_ws, size_t ws_size,
                              hipStream_t stream) {
    const float*     x_in    = (const float*)d_in[0];
    const long long* ei      = (const long long*)d_in[1];
    const long long* src     = ei;
    const long long* dst     = ei + N_EDGES;
    const float*     W_mlp   = (const float*)d_in[2];
    const float*     b_mlp   = (const float*)d_in[3];
    const float*     W_conv1 = (const float*)d_in[4];
    const float*     b_conv1 = (const float*)d_in[5];
    const float*     W_hid   = (const float*)d_in[6];
    const float*     b_hid   = (const float*)d_in[7];
    const float*     ln_g    = (const float*)d_in[8];
    const float*     ln_b    = (const float*)d_in[9];
    const float*     W_post  = (const float*)d_in[10];
    const float*     b_post  = (const float*)d_in[11];
    const float*     W_lin   = (const float*)d_in[12];
    const float*     b_lin   = (const float*)d_in[13];

    char* ws = (char*)d_ws;
    const size_t featBytes = (size_t)N_NODES * HID * sizeof(float);   // 51.2 MB
    float* bufP = (float*)(ws);
    float* bufQ = (float*)(ws + featBytes);
    float* bufR = (float*)(ws + 2 * featBytes);
    float* dinv = (float*)(ws + 3 * featBytes);

    const int MT = N_NODES / 16;                  // 3125 exact
    dim3 ggrid((MT + 3) / 4, 4);
    dim3 gblk(128);
    const int AGG4 = N_NODES * (HID / 4);         // float4 count of a feature buffer

    // degrees -> dinv (in place)
    zero_f4<<<(N_NODES / 4 + 255) / 256, 256, 0, stream>>>(dinv, N_NODES / 4);
    deg_count<<<(N_EDGES + 255) / 256, 256, 0, stream>>>(dst, dinv, N_EDGES);
    finish_dinv<<<(N_NODES + 255) / 256, 256, 0, stream>>>(dinv, N_NODES);

    // pre-MLP (bias+celu fused)
    gemm256_wmma<true><<<ggrid, gblk, 0, stream>>>(x_in, W_mlp, b_mlp, bufP, MT);
    gemm256_wmma<true><<<ggrid, gblk, 0, stream>>>(bufP, W_mlp + 65536, b_mlp + 256, bufQ, MT);

    // conv1
    gemm256_wmma<false><<<ggrid, gblk, 0, stream>>>(bufQ, W_conv1, nullptr, bufP, MT);
    zero_f4<<<(AGG4 + 255) / 256, 256, 0, stream>>>(bufR, AGG4);
    edge_scatter<<<(N_EDGES + 7) / 8, 256, 0, stream>>>(bufP, src, dst, dinv, bufR, N_EDGES);
    node_update<<<(N_NODES + 7) / 8, 256, 0, stream>>>(bufR, bufP, dinv, b_conv1, bufQ, N_NODES);

    // hidden convs + layernorm
    for (int i = 0; i < 3; ++i) {
        gemm256_wmma<false><<<ggrid, gblk, 0, stream>>>(bufQ, W_hid + i * 65536, nullptr, bufP, MT);
        zero_f4<<<(AGG4 + 255) / 256, 256, 0, stream>>>(bufR, AGG4);
        edge_scatter<<<(N_EDGES + 7) / 8, 256, 0, stream>>>(bufP, src, dst, dinv, bufR, N_EDGES);
        node_update_ln<<<(N_NODES + 7) / 8, 256, 0, stream>>>(
            bufR, bufP, dinv, b_hid + i * 256, ln_g, ln_b, bufQ, N_NODES);
    }

    // post-MLP (bias+celu fused)
    gemm256_wmma<true><<<ggrid, gblk, 0, stream>>>(bufQ, W_post, b_post, bufP, MT);
    gemm256_wmma<true><<<ggrid, gblk, 0, stream>>>(bufP, W_post + 65536, b_post + 256, bufQ, MT);

    // final projection + sigmoid
    final_lin<<<(N_NODES * 25 + 255) / 256, 256, 0, stream>>>(bufQ, W_lin, b_lin,
                                                              (float*)d_out, N_NODES);
}